// Scatter_Rendering_40235253629026
// MI455X (gfx1250) — compile-verified
//
#include <hip/hip_runtime.h>
#include <stdint.h>

// ---------------------------------------------------------------------------
// Depth-dependent scatter splatting (gather form), B=4, C=4(rgb+disp),
// H=W=1024, LENS=11 window, edge-clamped.
//
// CDNA5 strategy: async global->LDS halo staging (ASYNCcnt), float4-packed
// LDS tile so each tap is one ds_load_b128, fast v_exp_f32 / v_rcp_f32 for
// the sigmoid gate and final normalization.
// ---------------------------------------------------------------------------

#define LENS   11
#define RAD    5
#define TILE_W 32
#define TILE_H 8
#define HALO_W (TILE_W + 2 * RAD)   // 42
#define HALO_H (TILE_H + 2 * RAD)   // 18
#define HALO_P 44                   // padded pitch (float4 units)
#define NLOAD  (HALO_W * HALO_H)    // 756 elements per channel

#ifndef __has_builtin
#define __has_builtin(x) 0
#endif

#if __has_builtin(__builtin_amdgcn_global_load_async_to_lds_b32)
#define HAVE_ASYNC_BUILTIN 1
#else
#define HAVE_ASYNC_BUILTIN 0
#endif

#if __has_builtin(__builtin_amdgcn_s_wait_asynccnt)
#define HAVE_WAIT_BUILTIN 1
#else
#define HAVE_WAIT_BUILTIN 0
#endif

typedef __attribute__((address_space(1))) int g_int;   // global (prints as __device__ int*)
typedef __attribute__((address_space(3))) int l_int;   // LDS    (prints as __shared__ int*)

// One 4-byte async copy: global -> LDS. Tracked by ASYNCcnt.
__device__ __forceinline__ void g2l_async_b32(const float* g, float* l) {
#if HAVE_ASYNC_BUILTIN
  __builtin_amdgcn_global_load_async_to_lds_b32(
      (g_int*)g,            // addrspacecast generic -> AS1 (identity value)
      (l_int*)l,            // addrspacecast generic -> AS3 (LDS offset)
      /*offset=*/0, /*cpol=*/0);
#else
  unsigned loff = (unsigned)(uintptr_t)(l_int*)l;
  asm volatile("global_load_async_to_lds_b32 %0, %1, off"
               :: "v"(loff), "v"(g) : "memory");
#endif
}

__device__ __forceinline__ void wait_async0() {
#if HAVE_WAIT_BUILTIN
  __builtin_amdgcn_s_wait_asynccnt(0);
#else
  asm volatile("s_wait_asynccnt 0" ::: "memory");
#endif
}

__global__ __launch_bounds__(256, 4)
void dof_scatter_tile_kernel(const float* __restrict__ x,
                             const float* __restrict__ lens_effects,
                             const float* __restrict__ diskernel,
                             const float* __restrict__ lens_mask,
                             float* __restrict__ out,
                             int H, int W)
{
  // r,g,b,disp packed per pixel -> one ds_load_b128 per tap.
  __shared__ float4 s_tile[HALO_H][HALO_P];
  __shared__ float  s_dm[LENS * LENS];   // dist - 0.5 (pre-fused)
  __shared__ float  s_m [LENS * LENS];   // aperture mask

  const int b   = blockIdx.z;
  const int ty0 = blockIdx.y * TILE_H;
  const int tx0 = blockIdx.x * TILE_W;
  const int tid = threadIdx.y * TILE_W + threadIdx.x;   // 0..255

  const size_t plane = (size_t)H * (size_t)W;
  const float* xb = x + (size_t)b * 4u * plane;

  if (tid < LENS * LENS) {
    s_dm[tid] = diskernel[tid] - 0.5f;
    s_m [tid] = lens_mask[tid];
  }

  // ---- async halo staging: 4 channels x 756 elements, edge-clamped ----
  float* s_base = (float*)&s_tile[0][0];
  for (int idx = tid; idx < 4 * NLOAD; idx += 256) {
    const int c  = idx / NLOAD;            // channel 0..3
    const int e  = idx - c * NLOAD;        // element in halo
    const int ly = e / HALO_W;
    const int lx = e - ly * HALO_W;
    int gy = ty0 + ly - RAD;  gy = (gy < 0) ? 0 : ((gy > H - 1) ? H - 1 : gy);
    int gx = tx0 + lx - RAD;  gx = (gx < 0) ? 0 : ((gx > W - 1) ? W - 1 : gx);
    const float* g = xb + c * plane + (size_t)gy * W + gx;
    float* l = s_base + ((size_t)(ly * HALO_P + lx) * 4u + (unsigned)c);
    g2l_async_b32(g, l);
  }

  wait_async0();        // this wave's ASYNCcnt == 0 (LDS writes landed)
  __syncthreads();      // all waves' halo visible

  // ---- per-pixel accumulation over 121 taps ----
  const int lx = threadIdx.x;   // 0..31
  const int ly = threadIdx.y;   // 0..7
  const float scale = lens_effects[b];                 // per-batch, scalar
  const float dd    = s_tile[ly + RAD][lx + RAD].w;    // dest disparity

  float nr = 0.f, ng = 0.f, nb = 0.f, den = 0.f;
  for (int dy = 0; dy < LENS; ++dy) {
    const float4* row = &s_tile[ly + dy][lx];
    const float*  dmr = &s_dm[dy * LENS];
    const float*  mr  = &s_m [dy * LENS];
#pragma unroll
    for (int dx = 0; dx < LENS; ++dx) {
      const float4 s = row[dx];            // ds_load_b128: r,g,b,disp
      const float sd  = s.w;
      const float coc = fabsf(scale * sd);
      float t = coc - dmr[dx];                         // coc - dist + 0.5
      t = fminf(fmaxf(t, 0.0f), 1.0f);                 // clip
      float w = mr[dx] * t;
      const float e = __expf(-4.0f * (sd - dd));       // v_exp_f32
      w *= __builtin_amdgcn_rcpf(1.0f + e);            // sigmoid gate
      nr  += w * s.x;
      ng  += w * s.y;
      nb  += w * s.z;
      den += w;
    }
  }

  const int oy = ty0 + ly, ox = tx0 + lx;
  if (oy < H && ox < W) {
    const float inv = __builtin_amdgcn_rcpf(den + 1e-8f);
    float* ob = out + (size_t)b * 3u * plane + (size_t)oy * W + ox;
    ob[0]         = nr * inv;
    ob[plane]     = ng * inv;
    ob[2 * plane] = nb * inv;
  }
}

extern "C" void kernel_launch(void* const* d_in, const int* in_sizes, int n_in,
                              void* d_out, int out_size, void* d_ws, size_t ws_size,
                              hipStream_t stream) {
  const float* x  = (const float*)d_in[0];   // (B,4,H,W) fp32
  const float* le = (const float*)d_in[1];   // (B,1)     fp32
  const float* dk = (const float*)d_in[2];   // (11,11)   fp32
  const float* lm = (const float*)d_in[3];   // (11,11)   fp32
  float* out = (float*)d_out;                // (B,3,H,W) fp32

  const int B = in_sizes[1];                         // 4
  const int W = 1024;                                // fixed by setup_inputs
  const int H = (int)((size_t)in_sizes[0] / (4u * (size_t)B * (size_t)W));

  dim3 block(TILE_W, TILE_H, 1);
  dim3 grid((W + TILE_W - 1) / TILE_W, (H + TILE_H - 1) / TILE_H, B);
  hipLaunchKernelGGL(dof_scatter_tile_kernel, grid, block, 0, stream,
                     x, le, dk, lm, out, H, W);
}